// MyGraphPoolOut2D_56324201120447
// MI455X (gfx1250) — compile-verified
//
#include <hip/hip_runtime.h>

// Voxel grid max-pool scatter for MI455X (gfx1250, wave32).
// out[b*4096 + (px/4)*64 + (py/4)][c] = max over points; empty cells = 0.
// Memory-bound: ~520 MB traffic -> ~22 us at 23.3 TB/s. Output (64 MB) stays
// hot in the 192 MB L2 for the 64M native u32-max atomics; the once-streamed
// x array (256 MB) is loaded non-temporally so it doesn't evict it.
// Exact & deterministic via order-preserving u32 keys (max is associative).

#define GRIDD      64
#define NUM_CELLS  4096
#define CCH        64
#define POOL_SHIFT 2

typedef float vf4 __attribute__((ext_vector_type(4)));   // clang vector: NT-load legal

// Order-preserving map float -> uint32 (monotone; max over keys == key of max).
__device__ __forceinline__ unsigned f2key(float f) {
    unsigned b = __float_as_uint(f);
    return (b & 0x80000000u) ? ~b : (b | 0x80000000u);
}
__device__ __forceinline__ float key2f(unsigned k) {
    unsigned b = (k & 0x80000000u) ? (k & 0x7FFFFFFFu) : ~k;
    return __uint_as_float(b);
}

// ---- Pass 1: init output to sentinel 0u (below every real key) -------------
// Regular-temporal stores: keep sentinel lines resident in L2 for the atomics.
__global__ __launch_bounds__(256) void vp_init(unsigned* __restrict__ out, int n4) {
    const uint4 z = {0u, 0u, 0u, 0u};
    const int stride = gridDim.x * blockDim.x;
    for (int i = blockIdx.x * blockDim.x + threadIdx.x; i < n4; i += stride)
        reinterpret_cast<uint4*>(out)[i] = z;
}

// ---- Pass 2: scatter-max. 16 threads per point, vf4 per thread -------------
__global__ __launch_bounds__(256) void vp_scatter(const vf4*  __restrict__ x4,
                                                  const int2* __restrict__ pos2,
                                                  const int*  __restrict__ batch,
                                                  unsigned*   __restrict__ out,
                                                  int npts) {
    const int total  = npts * (CCH / 4);                  // 16M work items (fits i32)
    const int stride = gridDim.x * blockDim.x;
    for (int t = blockIdx.x * blockDim.x + threadIdx.x; t < total; t += stride) {
        // gfx1250: prefetch next tile of the dominant 256MB x-stream
        if (t + stride < total)
            __builtin_prefetch(&x4[t + stride], 0, 1);    // -> global_prefetch_b8

        const int i  = t >> 4;                            // point index
        const int c4 = t & 15;                            // float4 chunk within row

        // x is streamed exactly once: non-temporal b128 load (don't pollute L2)
        const vf4 v = __builtin_nontemporal_load(&x4[t]);

        // pos/batch loads are identical across 16 consecutive lanes (broadcast)
        const int2 p   = pos2[i];
        const int cell = batch[i] * NUM_CELLS
                       + (p.x >> POOL_SHIFT) * GRIDD
                       + (p.y >> POOL_SHIFT);

        unsigned* dst = out + cell * CCH + c4 * 4;        // < 2^31, int math
        atomicMax(dst + 0, f2key(v.x));                   // global_atomic_max_u32
        atomicMax(dst + 1, f2key(v.y));                   // (resolves in 192MB L2)
        atomicMax(dst + 2, f2key(v.z));
        atomicMax(dst + 3, f2key(v.w));
    }
}

// ---- Pass 3: invert keys in place; untouched cells -> 0.0f -----------------
// Loads hit L2 (output is hot); stores are last-touch -> non-temporal.
__global__ __launch_bounds__(256) void vp_final(unsigned* __restrict__ buf, int n) {
    const int stride = gridDim.x * blockDim.x;
    for (int i = blockIdx.x * blockDim.x + threadIdx.x; i < n; i += stride) {
        unsigned k = buf[i];
        float r = (k == 0u) ? 0.0f : key2f(k);
        __builtin_nontemporal_store(r, reinterpret_cast<float*>(buf) + i);
    }
}

extern "C" void kernel_launch(void* const* d_in, const int* in_sizes, int n_in,
                              void* d_out, int out_size, void* d_ws, size_t ws_size,
                              hipStream_t stream) {
    const float* x     = (const float*)d_in[0];   // [N, 64] f32
    const int*   pos   = (const int*)  d_in[1];   // [N, 2]  int
    const int*   batch = (const int*)  d_in[2];   // [N]     int (sorted)
    unsigned*    out_u = (unsigned*)   d_out;     // 16.78M elems: keys, then floats

    const int npts = in_sizes[2];                 // N = 1,000,000
    const int n4   = out_size / 4;                // uint4 count for init

    vp_init   <<<4096, 256, 0, stream>>>(out_u, n4);
    vp_scatter<<<8192, 256, 0, stream>>>((const vf4*)x, (const int2*)pos,
                                         batch, out_u, npts);
    vp_final  <<<8192, 256, 0, stream>>>(out_u, out_size);
}